// HorizontalWidthTokenPyramid_27324581937291
// MI455X (gfx1250) — compile-verified
//
#include <hip/hip_runtime.h>
#include <hip/hip_bf16.h>

#ifndef __has_builtin
#define __has_builtin(x) 0
#endif

#if __has_builtin(__builtin_amdgcn_global_load_async_to_lds_b128) && \
    __has_builtin(__builtin_amdgcn_s_wait_asynccnt)
#define HWTP_ASYNC 1
#else
#define HWTP_ASYNC 0
#endif

#define AS1 __attribute__((address_space(1)))
#define AS3 __attribute__((address_space(3)))

typedef float v4f __attribute__((ext_vector_type(4)));
typedef int   v4i __attribute__((ext_vector_type(4)));

namespace {
constexpr int kSlices      = 32 * 256 * 32;          // n*c*s = 262144
constexpr int kTileSlices  = 8;                      // slices per wave-tile
constexpr int kTiles       = kSlices / kTileSlices;  // 32768
constexpr int kOutPerSlice = 124;                    // 31 bins * 4 tokens
constexpr int kSliceBytes  = 1024;                   // 16*16 f32
constexpr int kLdsPitch    = 1056;                   // 1024 + 32B bank skew
constexpr int kLdsTile     = kTileSlices * kLdsPitch;
constexpr int kWavesPerBlk = 2;
constexpr int kBlock       = 32 * kWavesPerBlk;
constexpr int kGrid        = 4096;                   // 8192 waves -> 4 tiles each
}  // namespace

__device__ __forceinline__ float hsum4(v4f v) { return (v.x + v.y) + (v.z + v.w); }
__device__ __forceinline__ float hmax4(v4f v) {
  return fmaxf(fmaxf(v.x, v.y), fmaxf(v.z, v.w));
}
__device__ __forceinline__ v4f vmax4(v4f a, v4f b) {
  v4f r;
  r.x = fmaxf(a.x, b.x); r.y = fmaxf(a.y, b.y);
  r.z = fmaxf(a.z, b.z); r.w = fmaxf(a.w, b.w);
  return r;
}
// width pooling of one bin-row's 4 columns: mean + max
__device__ __forceinline__ float pool4(v4f bv) {
  return fmaf(hsum4(bv), 0.25f, hmax4(bv));
}

// Full row-pyramid + width pooling for one lane's 4 columns x 16 rows.
// tok[0..15]=b16 bins, [16..23]=b8, [24..27]=b4, [28..29]=b2, [30]=b1.
template <typename RowFn>
__device__ __forceinline__ void hwtp_compute(RowFn row, float tok[31]) {
  v4f s2[8], m2[8];
#pragma unroll
  for (int j = 0; j < 8; ++j) {
    v4f a = row(2 * j);
    v4f b = row(2 * j + 1);
    // b=16: bin value = mean+max over 1 row = 2*v; pool folds the 2x in.
    tok[2 * j]     = fmaf(hsum4(a), 0.5f, 2.0f * hmax4(a));
    tok[2 * j + 1] = fmaf(hsum4(b), 0.5f, 2.0f * hmax4(b));
    s2[j] = a + b;
    m2[j] = vmax4(a, b);
    tok[16 + j] = pool4(s2[j] * 0.5f + m2[j]);   // b=8
  }
  v4f s4[4], m4[4];
#pragma unroll
  for (int j = 0; j < 4; ++j) {
    s4[j] = s2[2 * j] + s2[2 * j + 1];
    m4[j] = vmax4(m2[2 * j], m2[2 * j + 1]);
    tok[24 + j] = pool4(s4[j] * 0.25f + m4[j]);  // b=4
  }
  v4f s8[2], m8[2];
#pragma unroll
  for (int j = 0; j < 2; ++j) {
    s8[j] = s4[2 * j] + s4[2 * j + 1];
    m8[j] = vmax4(m4[2 * j], m4[2 * j + 1]);
    tok[28 + j] = pool4(s8[j] * 0.125f + m8[j]); // b=2
  }
  v4f s16 = s8[0] + s8[1];
  v4f m16 = vmax4(m8[0], m8[1]);
  tok[30] = pool4(s16 * 0.0625f + m16);          // b=1
}

__global__ __launch_bounds__(kBlock)
void HorizontalWidthTokenPyramid_27324581937291_kernel(const float* __restrict__ x,
                                                       float* __restrict__ out) {
  const int lane   = threadIdx.x & 31;
  const int wave   = threadIdx.x >> 5;
  const int wgid   = blockIdx.x * kWavesPerBlk + wave;
  const int nwaves = gridDim.x * kWavesPerBlk;
  const int sl = lane >> 2;  // slice within tile (0..7)
  const int t  = lane & 3;   // width token     (0..3)

#if HWTP_ASYNC
  // Per-wave double-buffered LDS staging, slices padded to skew banks.
  __shared__ __align__(16) unsigned char smem[kWavesPerBlk * 2 * kLdsTile];
  AS3 unsigned char* lbase =
      (AS3 unsigned char*)smem + (unsigned)wave * (2u * kLdsTile);

  auto issue_tile = [&](int tile, int buf) {
    AS1 unsigned char* g = (AS1 unsigned char*)x +
                           (size_t)tile * (kTileSlices * kSliceBytes) + lane * 16;
    AS3 unsigned char* l = lbase + buf * kLdsTile + lane * 16;
#pragma unroll
    for (int s = 0; s < kTileSlices; ++s) {
#pragma unroll
      for (int h = 0; h < 2; ++h) {  // 2 x 512B per slice
        __builtin_amdgcn_global_load_async_to_lds_b128(
            (AS1 v4i*)(g + s * kSliceBytes + h * 512),
            (AS3 v4i*)(l + s * kLdsPitch + h * 512),
            0, 0);
      }
    }
  };

  int tile = wgid;
  if (tile < kTiles) issue_tile(tile, 0);
  int cur = 0;
  for (; tile < kTiles; tile += nwaves) {
    const int nxt = tile + nwaves;
    if (nxt < kTiles) {
      issue_tile(nxt, cur ^ 1);
      __builtin_amdgcn_s_wait_asynccnt(16);  // current tile's 16 copies done
    } else {
      __builtin_amdgcn_s_wait_asynccnt(0);
    }
    AS3 const v4f* rows = (AS3 const v4f*)(lbase + cur * kLdsTile +
                                           sl * kLdsPitch + t * 16);
    float tok[31];
    hwtp_compute([&](int r) -> v4f { return rows[r * 4]; }, tok);  // r*64 bytes

    float* op = out + (size_t)tile * (kTileSlices * kOutPerSlice) +
                sl * kOutPerSlice + t;
#pragma unroll
    for (int j = 0; j < 31; ++j) op[4 * j] = tok[j];
    cur ^= 1;
  }
#else
  // Fallback: direct b128 loads + gfx1250 global_prefetch of next tile.
  for (int tile = wgid; tile < kTiles; tile += nwaves) {
    const float* base = x + (size_t)tile * (kTileSlices * 256) + sl * 256 + t * 4;
    if (tile + nwaves < kTiles)
      __builtin_prefetch(base + (size_t)nwaves * (kTileSlices * 256), 0, 1);
    float tok[31];
    hwtp_compute([&](int r) -> v4f {
      return *(const v4f*)(base + r * 16);
    }, tok);
    float* op = out + (size_t)tile * (kTileSlices * kOutPerSlice) +
                sl * kOutPerSlice + t;
#pragma unroll
    for (int j = 0; j < 31; ++j) op[4 * j] = tok[j];
  }
#endif
}

extern "C" void kernel_launch(void* const* d_in, const int* in_sizes, int n_in,
                              void* d_out, int out_size, void* d_ws, size_t ws_size,
                              hipStream_t stream) {
  const float* x = (const float*)d_in[0];
  float* out = (float*)d_out;
  (void)in_sizes; (void)n_in; (void)out_size; (void)d_ws; (void)ws_size;
  hipLaunchKernelGGL(HorizontalWidthTokenPyramid_27324581937291_kernel,
                     dim3(kGrid), dim3(kBlock), 0, stream, x, out);
}